// Net1_25142738550810
// MI455X (gfx1250) — compile-verified
//
#include <hip/hip_runtime.h>

typedef __attribute__((ext_vector_type(2))) float v2f;
typedef __attribute__((ext_vector_type(8))) float v8f;

#define F_IN 96
#define DIM  128

// ---------------------------------------------------------------------------
// Zero the aggregation buffer (grid-stride).
// ---------------------------------------------------------------------------
__global__ void zero_kernel(float* __restrict__ p, long long n) {
  long long i = (long long)blockIdx.x * blockDim.x + threadIdx.x;
  long long stride = (long long)gridDim.x * blockDim.x;
  for (; i < n; i += stride) p[i] = 0.0f;
}

// ---------------------------------------------------------------------------
// Edge phase: msg = relu(x[src] + edge_attr); atomic scatter-add into agg[dst].
// One thread per (edge, float4 chunk): 24 chunks per edge (F_IN = 96).
// edge_attr loads fully coalesced; x gathers are L2-resident (x = 19 MB << 192 MB L2).
// ---------------------------------------------------------------------------
__global__ void edge_scatter_kernel(const float* __restrict__ x,
                                    const int*   __restrict__ edge_index, // [2, E]
                                    const float* __restrict__ edge_attr,  // [E, F_IN]
                                    float* __restrict__ agg,              // [N, F_IN]
                                    int E, long long nWork) {
  long long gid = (long long)blockIdx.x * blockDim.x + threadIdx.x;
  if (gid >= nWork) return;
  int e = (int)(gid / (F_IN / 4));
  int c = (int)(gid % (F_IN / 4)) * 4;
  int s = edge_index[e];      // src
  int d = edge_index[E + e];  // dst
  const float4 xv = *(const float4*)(x + (long long)s * F_IN + c);
  const float4 av = *(const float4*)(edge_attr + (long long)e * F_IN + c);
  float4 m;
  m.x = fmaxf(xv.x + av.x, 0.0f);
  m.y = fmaxf(xv.y + av.y, 0.0f);
  m.z = fmaxf(xv.z + av.z, 0.0f);
  m.w = fmaxf(xv.w + av.w, 0.0f);
  float* dstp = agg + (long long)d * F_IN + c;
  atomicAdd(dstp + 0, m.x);
  atomicAdd(dstp + 1, m.y);
  atomicAdd(dstp + 2, m.z);
  atomicAdd(dstp + 3, m.w);
}

// ---------------------------------------------------------------------------
// Fused GEMM: out[r, 0:128] = act( in[r, 0:K] @ W[K,128] + bias ), act = relu.
// FUSE variant computes in = A + (1+eps)*X on the fly while filling LDS (GIN
// self-term fused into GEMM1).
//
// Block = 256 threads = 8 wave32; block owns 16 rows; wave w owns cols
// [16w, 16w+16). K-loop of V_WMMA_F32_16X16X4_F32 (fp32 exact, matches ref).
//
// Fragment layouts per CDNA5 ISA:
//   A (16x4 f32):  lanes 0-15 -> M=lane, {K=k0,k0+1}; lanes 16-31 -> {k0+2,k0+3}
//   B (4x16 f32):  row-striped, mirror of A
//   C/D (16x16):   8 VGPRs; lanes 0-15 -> M=i, lanes 16-31 -> M=8+i; N=lane&15
// LDS tile padded to stride K+4 -> conflict-free A reads for both K=96, K=128.
// ---------------------------------------------------------------------------
template <int K, bool FUSE, bool RELU>
__global__ void gemm_relu_wmma(const float* __restrict__ A,      // [nRows, K]
                               const float* __restrict__ X,      // [nRows, K] (FUSE)
                               const float* __restrict__ epsPtr, // [1] (FUSE)
                               const float* __restrict__ W,      // [K, DIM]
                               const float* __restrict__ bias,   // [DIM]
                               float* __restrict__ out,          // [nRows, DIM]
                               int nRows) {
  constexpr int KP = K + 4;
  __shared__ float tile[16 * KP];

  const int rowBase = blockIdx.x * 16;
  const float scale = FUSE ? (1.0f + epsPtr[0]) : 0.0f;

  // Cooperative staged load: 16 rows x K floats as float4 (coalesced).
  constexpr int NF4 = 16 * (K / 4);
  for (int f = threadIdx.x; f < NF4; f += blockDim.x) {
    int r   = f / (K / 4);
    int off = (f % (K / 4)) * 4;
    int gr  = rowBase + r;
    if (gr >= nRows) gr = nRows - 1;  // safe clamp (N is a multiple of 16 here)
    long long gidx = (long long)gr * K + off;
    float4 v = *(const float4*)(A + gidx);
    if (FUSE) {
      float4 xv = *(const float4*)(X + gidx);
      v.x += scale * xv.x;
      v.y += scale * xv.y;
      v.z += scale * xv.z;
      v.w += scale * xv.w;
    }
    *(float4*)(tile + r * KP + off) = v;
  }
  __syncthreads();

  const int lane = threadIdx.x & 31;
  const int wave = threadIdx.x >> 5;
  const int half = lane >> 4;   // 0: lanes 0-15, 1: lanes 16-31
  const int m    = lane & 15;
  const int col  = wave * 16 + m;

  v8f acc = {};
#pragma unroll
  for (int k0 = 0; k0 < K; k0 += 4) {
    const int kk = k0 + half * 2;
    v2f a, b;
    a.x = tile[m * KP + kk];
    a.y = tile[m * KP + kk + 1];
    b.x = W[kk * DIM + col];
    b.y = W[(kk + 1) * DIM + col];
    acc = __builtin_amdgcn_wmma_f32_16x16x4_f32(
        /*neg_a=*/false, a, /*neg_b=*/false, b,
        /*c_mod=*/(short)0, acc, /*reuse_a=*/false, /*reuse_b=*/false);
  }

  const float bc = bias[col];
#pragma unroll
  for (int i = 0; i < 8; ++i) {
    int r = rowBase + half * 8 + i;
    if (r < nRows) {
      float v = acc[i] + bc;
      if (RELU) v = fmaxf(v, 0.0f);
      out[(long long)r * DIM + col] = v;
    }
  }
}

// ---------------------------------------------------------------------------
// Final projection: y[r] = dot(h[r, 0:128], W2) + b2. One wave32 per row,
// 4 elements per lane, shuffle tree reduction.
// ---------------------------------------------------------------------------
__global__ void final_lin_kernel(const float* __restrict__ h,  // [nRows, DIM]
                                 const float* __restrict__ W2, // [DIM]
                                 const float* __restrict__ b2, // [1]
                                 float* __restrict__ out,      // [nRows]
                                 int nRows) {
  int row  = (int)(((long long)blockIdx.x * blockDim.x + threadIdx.x) >> 5);
  int lane = threadIdx.x & 31;
  if (row >= nRows) return;
  float s = 0.0f;
#pragma unroll
  for (int j = 0; j < 4; ++j) {
    int c = lane + j * 32;
    s += h[(long long)row * DIM + c] * W2[c];
  }
#pragma unroll
  for (int off = 16; off > 0; off >>= 1) s += __shfl_down(s, off, 32);
  if (lane == 0) out[row] = s + b2[0];
}

// ---------------------------------------------------------------------------
extern "C" void kernel_launch(void* const* d_in, const int* in_sizes, int n_in,
                              void* d_out, int out_size, void* d_ws, size_t ws_size,
                              hipStream_t stream) {
  const float* x    = (const float*)d_in[0];
  const int*   ei   = (const int*)d_in[1];   // [2, E]
  const float* ea   = (const float*)d_in[2]; // [E, F_IN]
  const float* eps  = (const float*)d_in[3]; // [1]
  const float* W_nn = (const float*)d_in[4]; // [F_IN, DIM]
  const float* b_nn = (const float*)d_in[5]; // [DIM]
  const float* W1   = (const float*)d_in[6]; // [DIM, DIM]
  const float* b1   = (const float*)d_in[7]; // [DIM]
  const float* W2   = (const float*)d_in[8]; // [DIM, 1]
  const float* b2   = (const float*)d_in[9]; // [1]

  const int N = in_sizes[0] / F_IN; // 50000
  const int E = in_sizes[2] / F_IN; // 800000

  // Workspace: region0 [N*DIM] = agg (first N*F_IN used) then reused as h2;
  //            region1 [N*DIM] = h1. Total 2*N*DIM*4 = 51.2 MB.
  float* ws  = (float*)d_ws;
  float* agg = ws;
  float* h1  = ws + (size_t)N * DIM;
  float* h2  = ws; // agg dead after GEMM1 -> reuse
  float* out = (float*)d_out;

  // 1) zero agg
  zero_kernel<<<1024, 256, 0, stream>>>(agg, (long long)N * F_IN);

  // 2) edge message + scatter-add
  long long nWork = (long long)E * (F_IN / 4);
  long long blocks = (nWork + 255) / 256;
  edge_scatter_kernel<<<(int)blocks, 256, 0, stream>>>(x, ei, ea, agg, E, nWork);

  // 3) GEMM1 fused with GIN self-term: h1 = relu((agg + (1+eps)x) @ W_nn + b_nn)
  int gBlocks = (N + 15) / 16;
  gemm_relu_wmma<F_IN, true, true><<<gBlocks, 256, 0, stream>>>(
      agg, x, eps, W_nn, b_nn, h1, N);

  // 4) GEMM2: h2 = relu(h1 @ W1 + b1)
  gemm_relu_wmma<DIM, false, true><<<gBlocks, 256, 0, stream>>>(
      h1, nullptr, nullptr, W1, b1, h2, N);

  // 5) final projection: out = h2 @ W2 + b2
  int fBlocks = (N * 32 + 255) / 256;
  final_lin_kernel<<<fBlocks, 256, 0, stream>>>(h2, W2, b2, out, N);
}